// GCN_80917183856728
// MI455X (gfx1250) — compile-verified
//
#include <hip/hip_runtime.h>

// ---------------------------------------------------------------------------
// GCN block for MI455X (gfx1250, wave32), fused two-pass version.
//   Z = [Ww ; Wrb] @ x     (256x64 channel GEMM, V_WMMA_F32_16X16X4_F32)
//   y = BN(Z[0:128] @ M),  r = BN(Z[128:256]),  out = relu(y + r)
// Roofline: ~17 GFLOP vs 23.3 TB/s HBM -> memory bound. K=64 makes the GEMM
// cheap to RECOMPUTE, so we never materialize Z (saves ~1.4 GB of traffic):
//   pass1: GEMM tile in LDS -> per-channel sum/sumsq (sliced atomics)
//   pass2: GEMM tile in LDS -> fused BN affine + residual + ReLU -> out
// Total HBM traffic ~492 MB (~21 us), fp32 WMMA keeps exact numerics.
// ---------------------------------------------------------------------------

typedef __attribute__((ext_vector_type(2))) float v2f;
typedef __attribute__((ext_vector_type(8))) float v8f;

#define NBATCH 64
#define CIN    64
#define COUT   128
#define NCH    256                  // stacked output channels (Ww rows ++ Wrb rows)
#define TT     300
#define VV     25
#define TVL    (TT * VV)            // 7500
#define EPSBN  1e-5f

#define NSLICE 64                   // stats contention slices
#define S2     42                   // xs LDS stride: 2*S2 % 64 = 20 -> split-K halves disjoint
#define S3     27                   // zs LDS stride: 8*S3 % 64 = 24 -> row halves disjoint

// ws layout (floats): stats[NSLICE*512] | scale_shift[512]
#define STATS_FLOATS (NSLICE * 512)

// ---------------------------------------------------------------------------
__global__ __launch_bounds__(512) void gcn_zero(float* p, int n) {
  int i = blockIdx.x * blockDim.x + threadIdx.x;
  if (i < n) p[i] = 0.0f;
}

// ---------------------------------------------------------------------------
// Block-cooperative GEMM of one (n,t) tile: zs[256][25(+pad)] = W @ x-tile + b.
// 8 waves; wave w owns rows [32w,32w+32) as 2 row-tiles x 2 col-tiles of WMMA.
// fp32 WMMA 16x16x4 fragments (ISA 7.12.2):
//   A: lanes 0-15 K={0,1}, lanes 16-31 K={2,3}, M = lane%16 (B mirrors, N=lane%16)
//   C/D: VGPR r -> M = r + (lane>=16 ? 8 : 0), N = lane%16
// ---------------------------------------------------------------------------
__device__ __forceinline__ void gcn_gemm_tile(
    const float* __restrict__ x,
    const float* __restrict__ Ww,  const float* __restrict__ bw,
    const float* __restrict__ Wrb, const float* __restrict__ brb,
    int n, int t, float* xs, float* zs)
{
  const int tid  = threadIdx.x;
  const int lane = tid & 31;
  const int wave = tid >> 5;

  // Stage x tile: 64 channels x 32 columns (25 valid, rest zero-padded).
  const float* xbase = x + ((long)n * CIN) * TVL + (long)t * VV;
  #pragma unroll
  for (int i = 0; i < 8; ++i) {
    int idx = tid + i * 256;          // 0..2047
    int c = idx >> 5;                 // channel 0..63
    int j = idx & 31;                 // padded column
    float v = 0.f;
    if (j < VV) v = xbase[(long)c * TVL + j];
    xs[c * S2 + j] = v;
  }
  __syncthreads();

  const int rowbase = wave * 32;
  const float* __restrict__ W    = (rowbase < COUT) ? Ww : Wrb;
  const float* __restrict__ bvec = (rowbase < COUT) ? bw : brb;
  const int rowoff = (rowbase < COUT) ? rowbase : (rowbase - COUT);

  const int mrow = lane & 15;
  const int kb   = (lane >> 4) << 1;

  v8f acc[2][2];
  {
    const v8f zero = {0.f,0.f,0.f,0.f,0.f,0.f,0.f,0.f};
    acc[0][0] = zero; acc[0][1] = zero; acc[1][0] = zero; acc[1][1] = zero;
  }

  for (int k0 = 0; k0 < CIN; k0 += 4) {
    v2f a0, a1;
    const float* wr0 = W + (long)(rowoff + mrow) * CIN + (k0 + kb);
    const float* wr1 = W + (long)(rowoff + 16 + mrow) * CIN + (k0 + kb);
    a0.x = wr0[0]; a0.y = wr0[1];
    a1.x = wr1[0]; a1.y = wr1[1];
    #pragma unroll
    for (int jt = 0; jt < 2; ++jt) {
      v2f b;
      const float* bs = &xs[(k0 + kb) * S2 + jt * 16 + mrow];
      b.x = bs[0];
      b.y = bs[S2];
      acc[0][jt] = __builtin_amdgcn_wmma_f32_16x16x4_f32(
          false, a0, false, b, (short)0, acc[0][jt], false, false);
      acc[1][jt] = __builtin_amdgcn_wmma_f32_16x16x4_f32(
          false, a1, false, b, (short)0, acc[1][jt], false, false);
    }
  }

  // Scatter z tile (+bias) to LDS; padded columns dropped.
  const int rowAdd = (lane >> 4) << 3;        // 0 or 8
  #pragma unroll
  for (int jt = 0; jt < 2; ++jt) {
    const int col = jt * 16 + mrow;
    if (col < VV) {
      #pragma unroll
      for (int rt = 0; rt < 2; ++rt) {
        const int row0  = rowbase + rt * 16 + rowAdd;
        const int boff0 = rowoff  + rt * 16 + rowAdd;
        #pragma unroll
        for (int r = 0; r < 8; ++r)
          zs[(row0 + r) * S3 + col] = acc[rt][jt][r] + bvec[boff0 + r];
      }
    }
  }
  __syncthreads();
}

// ---------------------------------------------------------------------------
// Pass 1: GEMM tile + per-channel sum/sumsq of pre-BN values.
// Thread o (0..255) owns stacked channel o of this tile.
// ---------------------------------------------------------------------------
__global__ __launch_bounds__(256) void gcn_pass1(
    const float* __restrict__ x,
    const float* __restrict__ Ww,  const float* __restrict__ bw,
    const float* __restrict__ Wrb, const float* __restrict__ brb,
    const float* __restrict__ Bm,
    float* __restrict__ stats)
{
  __shared__ float xs[CIN * S2];
  __shared__ float zs[NCH * S3];
  __shared__ float Ms[VV * 26];

  const int b = blockIdx.x;           // (n,t) tile, 0..19199
  const int n = b / TT;
  const int t = b - n * TT;
  const int tid = threadIdx.x;

  for (int i = tid; i < VV * VV; i += 256)
    Ms[(i / VV) * 26 + (i % VV)] = Bm[i] + 0.04f;   // M = D^-1/2 A D^-1/2 + B

  gcn_gemm_tile(x, Ww, bw, Wrb, brb, n, t, xs, zs); // syncs cover Ms too

  const int o = tid;
  float zv[VV];
  #pragma unroll
  for (int w = 0; w < VV; ++w) zv[w] = zs[o * S3 + w];

  float sum = 0.f, sq = 0.f;
  if (o < COUT) {
    for (int v = 0; v < VV; ++v) {
      float y = 0.f;
      #pragma unroll
      for (int w = 0; w < VV; ++w) y = fmaf(zv[w], Ms[w * 26 + v], y);
      sum += y; sq = fmaf(y, y, sq);
    }
  } else {
    #pragma unroll
    for (int v = 0; v < VV; ++v) { sum += zv[v]; sq = fmaf(zv[v], zv[v], sq); }
  }

  float* st = stats + (long)(b & (NSLICE - 1)) * 512;
  atomicAdd(&st[o], sum);
  atomicAdd(&st[NCH + o], sq);
}

// ---------------------------------------------------------------------------
// Reduce slices, fold mean/var + gamma/beta into per-channel scale & shift.
// ---------------------------------------------------------------------------
__global__ __launch_bounds__(256) void gcn_finalize(
    const float* __restrict__ stats,
    const float* __restrict__ gamma,    const float* __restrict__ beta,
    const float* __restrict__ gamma_rb, const float* __restrict__ beta_rb,
    float* __restrict__ ss)
{
  const int o = threadIdx.x;            // 0..255
  float s1 = 0.f, s2 = 0.f;
  for (int s = 0; s < NSLICE; ++s) {
    s1 += stats[s * 512 + o];
    s2 += stats[s * 512 + NCH + o];
  }
  const float cnt  = (float)((long)NBATCH * TT * VV);    // 480000
  const float mean = s1 / cnt;
  const float var  = s2 / cnt - mean * mean;             // biased (torch BN)
  const float g = (o < COUT) ? gamma[o] : gamma_rb[o - COUT];
  const float bb = (o < COUT) ? beta[o] : beta_rb[o - COUT];
  const float sc = g * rsqrtf(var + EPSBN);
  ss[o]       = sc;
  ss[NCH + o] = bb - mean * sc;
}

// ---------------------------------------------------------------------------
// Pass 2: recompute GEMM tile, fuse BN affines + residual + ReLU, write out.
// Thread pair (2*oc+vh) owns output channel oc, v-range half vh.
// ---------------------------------------------------------------------------
__global__ __launch_bounds__(256) void gcn_pass2(
    const float* __restrict__ x,
    const float* __restrict__ Ww,  const float* __restrict__ bw,
    const float* __restrict__ Wrb, const float* __restrict__ brb,
    const float* __restrict__ Bm,
    const float* __restrict__ ss,
    float* __restrict__ out)
{
  __shared__ float xs[CIN * S2];
  __shared__ float zs[NCH * S3];
  __shared__ float Ms[VV * 26];

  const int b = blockIdx.x;
  const int n = b / TT;
  const int t = b - n * TT;
  const int tid = threadIdx.x;

  for (int i = tid; i < VV * VV; i += 256)
    Ms[(i / VV) * 26 + (i % VV)] = Bm[i] + 0.04f;

  gcn_gemm_tile(x, Ww, bw, Wrb, brb, n, t, xs, zs);

  const int oc = tid >> 1;             // 0..127
  const int vh = tid & 1;
  const int v0 = vh ? 13 : 0;
  const int v1 = vh ? VV : 13;

  float zv[VV];
  #pragma unroll
  for (int w = 0; w < VV; ++w) zv[w] = zs[oc * S3 + w];

  const float sy = ss[oc],        ty = ss[NCH + oc];
  const float sr = ss[COUT + oc], tr = ss[NCH + COUT + oc];

  float* orow = out + (((long)n * COUT + oc) * TT + t) * VV;
  for (int v = v0; v < v1; ++v) {
    float y = 0.f;
    #pragma unroll
    for (int w = 0; w < VV; ++w) y = fmaf(zv[w], Ms[w * 26 + v], y);
    const float rv = zs[(COUT + oc) * S3 + v];
    orow[v] = fmaxf(fmaf(sy, y, ty) + fmaf(sr, rv, tr), 0.f);
  }
}

// ---------------------------------------------------------------------------
extern "C" void kernel_launch(void* const* d_in, const int* in_sizes, int n_in,
                              void* d_out, int out_size, void* d_ws, size_t ws_size,
                              hipStream_t stream) {
  const float* x        = (const float*)d_in[0];
  const float* Bm       = (const float*)d_in[1];
  const float* Ww       = (const float*)d_in[2];
  const float* bw       = (const float*)d_in[3];
  const float* gamma    = (const float*)d_in[4];
  const float* beta     = (const float*)d_in[5];
  const float* Wrb      = (const float*)d_in[6];
  const float* brb      = (const float*)d_in[7];
  const float* gamma_rb = (const float*)d_in[8];
  const float* beta_rb  = (const float*)d_in[9];

  float* stats = (float*)d_ws;
  float* ss    = stats + STATS_FLOATS;
  float* out   = (float*)d_out;

  const int ntile = NBATCH * TT;       // 19200 (n,t) tiles

  gcn_zero<<<(STATS_FLOATS + 511) / 512, 512, 0, stream>>>(stats, STATS_FLOATS);
  gcn_pass1<<<ntile, 256, 0, stream>>>(x, Ww, bw, Wrb, brb, Bm, stats);
  gcn_finalize<<<1, 256, 0, stream>>>(stats, gamma, beta, gamma_rb, beta_rb, ss);
  gcn_pass2<<<ntile, 256, 0, stream>>>(x, Ww, bw, Wrb, brb, Bm, ss, out);
}